// GraphConvolution_30889404793465
// MI455X (gfx1250) — compile-verified
//
#include <hip/hip_runtime.h>
#include <hip/hip_bf16.h>

// ---------------------------------------------------------------------------
// GraphConvolution on MI455X (gfx1250, wave32):
//   h   = X @ W                    (fp32 WMMA 16x16x4, 64-row blocks)
//   agg = scatter-add over edges   (global_atomic_add_f32, L2-resident)
//   out = relu(agg)
// ---------------------------------------------------------------------------

typedef __attribute__((ext_vector_type(2))) float v2f;
typedef __attribute__((ext_vector_type(8))) float v8f;
typedef __attribute__((ext_vector_type(4))) int   v4i;

#define IN_DIM  256
#define OUT_DIM 128
#define BM      64          // rows per GEMM block

// ---- optional CDNA5 async global->LDS staging (ASYNCcnt path) -------------
#if defined(__has_builtin)
#if __has_builtin(__builtin_amdgcn_global_load_async_to_lds_b128) && \
    __has_builtin(__builtin_amdgcn_s_wait_asynccnt)
#define HAVE_ASYNC_LDS 1
#endif
#endif
#ifndef HAVE_ASYNC_LDS
#define HAVE_ASYNC_LDS 0
#endif

#if HAVE_ASYNC_LDS
// Builtin signature (from hipcc diagnostic): b128 variant takes int4-vector
// pointers; arg0 = global (addrspace 1) source, arg1 = LDS (addrspace 3) dest.
typedef __attribute__((address_space(1))) v4i GlobalV4i;
typedef __attribute__((address_space(3))) v4i SharedV4i;
#endif

// ---------------------------------------------------------------------------
// Kernel 0: zero the accumulator buffer (d_out is poisoned before timing).
// ---------------------------------------------------------------------------
__global__ __launch_bounds__(256) void zero_f32(float* __restrict__ p, int n4) {
  int i = blockIdx.x * blockDim.x + threadIdx.x;
  if (i < n4) {
    ((float4*)p)[i] = make_float4(0.f, 0.f, 0.f, 0.f);
  }
}

// ---------------------------------------------------------------------------
// Kernel 1: dense GEMM H = X @ W with V_WMMA_F32_16X16X4_F32.
//   Block = 64 rows x 128 cols, 8 wave32s. Wave w owns col tile w*16 and all
//   four 16-row sub-tiles (4 x v8f accumulators) -> each B fragment is reused
//   by 4 WMMAs, cutting W re-reads from L2 by 4x vs a 16-row block.
//   A tile (64x256 f32 = 64 KB) staged in LDS; async global->LDS DMA when
//   available (full tiles), coalesced float4 fallback otherwise. Padded row
//   stride (260 floats) keeps per-lane LDS reads bank-conflict-free.
//
//   f32 16x16x4 VGPR layout (ISA 7.12.2):
//     A: lanes 0-15 -> M=lane, {K0,K1}; lanes 16-31 -> M=lane-16, {K2,K3}
//     C/D: VGPR i, lanes 0-15 -> M=i, N=lane; lanes 16-31 -> M=8+i, N=lane-16
// ---------------------------------------------------------------------------
__global__ __launch_bounds__(256) void gemm_wmma_f32(
    const float* __restrict__ X, const float* __restrict__ W,
    float* __restrict__ H, int nNodes) {
  __shared__ __align__(16) float As[BM][260];   // 66.5 KB

  const int r0   = blockIdx.x * BM;
  const int t    = threadIdx.x;
  const bool full = (r0 + BM) <= nNodes;

  // Stage A-tile: 64x256 floats = 4096 float4s, 16 per thread, coalesced.
  if (full) {
#if HAVE_ASYNC_LDS
    #pragma unroll
    for (int i = 0; i < 16; ++i) {
      const int f   = t + i * 256;
      const int row = f >> 6;
      const int cv  = f & 63;
      __builtin_amdgcn_global_load_async_to_lds_b128(
          (GlobalV4i*)(X + (size_t)(r0 + row) * IN_DIM + cv * 4),
          (SharedV4i*)&As[row][cv * 4], 0, 0);
    }
    __builtin_amdgcn_s_wait_asynccnt(0);
#else
    #pragma unroll
    for (int i = 0; i < 16; ++i) {
      const int f   = t + i * 256;
      const int row = f >> 6;
      const int cv  = f & 63;
      *(float4*)&As[row][cv * 4] =
          ((const float4*)(X + (size_t)(r0 + row) * IN_DIM))[cv];
    }
#endif
  } else {
    // Tail block: guarded loads with zero fill.
    #pragma unroll
    for (int i = 0; i < 16; ++i) {
      const int f   = t + i * 256;
      const int row = f >> 6;
      const int cv  = f & 63;
      float4 v = make_float4(0.f, 0.f, 0.f, 0.f);
      if (r0 + row < nNodes) {
        v = ((const float4*)(X + (size_t)(r0 + row) * IN_DIM))[cv];
      }
      *(float4*)&As[row][cv * 4] = v;
    }
  }
  __syncthreads();

  const int wave = t >> 5;         // 0..7 -> N tile
  const int lane = t & 31;
  const int col0 = wave * 16;
  const int m    = lane & 15;
  const int hi   = lane >> 4;      // half-wave select (K pair)

  v8f acc[4] = {};
  #pragma unroll 2
  for (int kk = 0; kk < IN_DIM / 4; ++kk) {
    const int k0 = kk * 4;
    v2f b;
    b.x = W[(size_t)(k0 + 2 * hi + 0) * OUT_DIM + col0 + m];
    b.y = W[(size_t)(k0 + 2 * hi + 1) * OUT_DIM + col0 + m];
    #pragma unroll
    for (int s = 0; s < 4; ++s) {
      v2f a;
      a.x = As[s * 16 + m][k0 + 2 * hi + 0];
      a.y = As[s * 16 + m][k0 + 2 * hi + 1];
      // 8 args: (neg_a, A, neg_b, B, c_mod, C, reuse_a, reuse_b)
      acc[s] = __builtin_amdgcn_wmma_f32_16x16x4_f32(
          false, a, false, b, (short)0, acc[s], false, false);
    }
  }

  // Scatter D tiles: sub-tile s, VGPR i -> row (s*16 + hi*8 + i), col0 + m.
  if (full) {
    #pragma unroll
    for (int s = 0; s < 4; ++s) {
      #pragma unroll
      for (int i = 0; i < 8; ++i) {
        const int row = r0 + s * 16 + hi * 8 + i;
        H[(size_t)row * OUT_DIM + (col0 + m)] = acc[s][i];
      }
    }
  } else {
    #pragma unroll
    for (int s = 0; s < 4; ++s) {
      #pragma unroll
      for (int i = 0; i < 8; ++i) {
        const int row = r0 + s * 16 + hi * 8 + i;
        if (row < nNodes) {
          H[(size_t)row * OUT_DIM + (col0 + m)] = acc[s][i];
        }
      }
    }
  }
}

// ---------------------------------------------------------------------------
// Kernel 2: COO SpMM scatter. One wave32 per edge; lane l handles the
// float4 column slice [4l, 4l+4). Gather of h[src] is one coalesced 512 B
// wave read; the scatter uses non-returning f32 atomics that resolve in L2
// (h and out both fit in the 192 MB L2).
// ---------------------------------------------------------------------------
__global__ __launch_bounds__(256) void spmm_scatter(
    const float* __restrict__ H,
    const int* __restrict__ esrc, const int* __restrict__ edst,
    const float* __restrict__ eval, float* __restrict__ out, int nEdges) {
  const int tid  = blockIdx.x * blockDim.x + threadIdx.x;
  const int e    = tid >> 5;
  const int lane = tid & 31;
  if (e >= nEdges) return;

  const int   s = esrc[e];
  const int   d = edst[e];
  const float v = eval[e];

  const float4 hv = ((const float4*)(H + (size_t)s * OUT_DIM))[lane];
  float* o = out + (size_t)d * OUT_DIM + lane * 4;
  atomicAdd(o + 0, v * hv.x);
  atomicAdd(o + 1, v * hv.y);
  atomicAdd(o + 2, v * hv.z);
  atomicAdd(o + 3, v * hv.w);
}

// ---------------------------------------------------------------------------
// Kernel 3: in-place ReLU.
// ---------------------------------------------------------------------------
__global__ __launch_bounds__(256) void relu_f32(float* __restrict__ p, int n4) {
  int i = blockIdx.x * blockDim.x + threadIdx.x;
  if (i < n4) {
    float4 v = ((float4*)p)[i];
    v.x = fmaxf(v.x, 0.f);
    v.y = fmaxf(v.y, 0.f);
    v.z = fmaxf(v.z, 0.f);
    v.w = fmaxf(v.w, 0.f);
    ((float4*)p)[i] = v;
  }
}

// ---------------------------------------------------------------------------
// Launch: zero -> gemm(WMMA) -> spmm(atomics) -> relu, all on `stream`.
// H (nNodes * 128 f32 = 51.2 MB) lives in d_ws.
// ---------------------------------------------------------------------------
extern "C" void kernel_launch(void* const* d_in, const int* in_sizes, int n_in,
                              void* d_out, int out_size, void* d_ws, size_t ws_size,
                              hipStream_t stream) {
  const float* X    = (const float*)d_in[0];   // [N, 256]
  const float* W    = (const float*)d_in[1];   // [256, 128]
  const int*   esrc = (const int*)d_in[2];     // [E]
  const int*   edst = (const int*)d_in[3];     // [E]
  const float* eval = (const float*)d_in[4];   // [E]
  float*       out  = (float*)d_out;           // [N, 128]
  float*       H    = (float*)d_ws;            // [N, 128] scratch

  const int nNodes = in_sizes[0] / IN_DIM;
  const int nEdges = in_sizes[2];
  const int outN4  = (nNodes * OUT_DIM) / 4;

  const int zgrid = (outN4 + 255) / 256;
  zero_f32<<<zgrid, 256, 0, stream>>>(out, outN4);

  const int ggrid = (nNodes + BM - 1) / BM;
  gemm_wmma_f32<<<ggrid, 256, 0, stream>>>(X, W, H, nNodes);

  const long long sthreads = (long long)nEdges * 32;
  const int sgrid = (int)((sthreads + 255) / 256);
  spmm_scatter<<<sgrid, 256, 0, stream>>>(H, esrc, edst, eval, out, nEdges);

  relu_f32<<<zgrid, 256, 0, stream>>>(out, outN4);
}